// CaptionBase_36094905155875
// MI455X (gfx1250) — compile-verified
//
#include <hip/hip_runtime.h>
#include <hip/hip_bf16.h>

#define B_    4
#define P_    128
#define T_    32
#define E_    300
#define EPAD  304     // E padded to multiple of 16 (zero-filled)
#define F_    2048
#define H_    512
#define H3_   1536
#define V_    3433
#define VP_   3456    // V padded to multiple of 64
#define NROW  512     // B_*P_ rows, indexed r = p*B_ + b
#define STEPS 31
#define KCH   32      // logits K-chunk staged in LDS
#define NCH   (H_ / KCH)

typedef float v2f __attribute__((ext_vector_type(2)));
typedef float v8f __attribute__((ext_vector_type(8)));
typedef int   v4i __attribute__((vector_size(4 * sizeof(int))));

#if defined(__has_builtin)
#if __has_builtin(__builtin_amdgcn_global_load_async_to_lds_b128)
#define HAVE_ASYNC 1
#endif
#endif
#ifndef HAVE_ASYNC
#define HAVE_ASYNC 0
#endif

typedef __attribute__((address_space(1))) v4i g_v4i;   // global
typedef __attribute__((address_space(3))) v4i l_v4i;   // LDS

__device__ __forceinline__ void async_copy16(const float* g, float* l) {
#if HAVE_ASYNC
  __builtin_amdgcn_global_load_async_to_lds_b128((g_v4i*)const_cast<float*>(g),
                                                 (l_v4i*)l, 0, 0);
#else
  *(float4*)l = *(const float4*)g;
#endif
}

__device__ __forceinline__ void async_wait2() {
#if HAVE_ASYNC
#if __has_builtin(__builtin_amdgcn_s_wait_asynccnt)
  __builtin_amdgcn_s_wait_asynccnt(2);
#else
  asm volatile("s_wait_asynccnt 0x2" ::: "memory");
#endif
#endif
}

__device__ __forceinline__ void async_wait0() {
#if HAVE_ASYNC
#if __has_builtin(__builtin_amdgcn_s_wait_asynccnt)
  __builtin_amdgcn_s_wait_asynccnt(0);
#else
  asm volatile("s_wait_asynccnt 0x0" ::: "memory");
#endif
#endif
}

// V_WMMA_F32_16X16X4_F32 : D(16x16 f32) = A(16x4 f32) * B(4x16 f32) + C
__device__ __forceinline__ v8f wmma4(v2f a, v2f b, v8f c) {
  return __builtin_amdgcn_wmma_f32_16x16x4_f32(false, a, false, b, (short)0, c,
                                               false, false);
}

__device__ __forceinline__ float sigf(float v) {
  return 1.0f / (1.0f + expf(-v));
}

// order-preserving float key; ties prefer smaller column (first occurrence)
__device__ __forceinline__ unsigned long long pack_key(float f, int col) {
  unsigned int u = __float_as_uint(f);
  u = (u & 0x80000000u) ? ~u : (u | 0x80000000u);
  return ((unsigned long long)u << 32) | (unsigned int)(~(unsigned int)col);
}

__device__ __forceinline__ unsigned long long shfl_xor_u64(unsigned long long v, int m) {
  unsigned int lo = (unsigned int)v, hi = (unsigned int)(v >> 32);
  lo = (unsigned int)__shfl_xor((int)lo, m, 32);
  hi = (unsigned int)__shfl_xor((int)hi, m, 32);
  return ((unsigned long long)hi << 32) | lo;
}

// ---------------- prep kernels (once per launch) ----------------
// W_ih [3H,E] -> pair-interleaved K-major: WihT[(k>>1)*2*H3 + j*2 + (k&1)]
__global__ void prep_ih(const float* __restrict__ W_ih, float* __restrict__ WihT) {
  int g = blockIdx.x * blockDim.x + threadIdx.x;
  if (g >= EPAD * H3_) return;
  int pr = g / (2 * H3_), rem = g - pr * (2 * H3_);
  int j = rem >> 1, kl = rem & 1;
  int k = 2 * pr + kl;
  WihT[g] = (k < E_) ? W_ih[j * E_ + k] : 0.0f;
}

__global__ void prep_hh(const float* __restrict__ W_hh, float* __restrict__ WhhT) {
  int g = blockIdx.x * blockDim.x + threadIdx.x;
  if (g >= H_ * H3_) return;
  int pr = g / (2 * H3_), rem = g - pr * (2 * H3_);
  int j = rem >> 1, kl = rem & 1;
  int k = 2 * pr + kl;
  WhhT[g] = W_hh[j * H_ + k];
}

// W_cls [H,V] -> padded (VP_) pair-interleaved: Wp[(k>>1)*2*VP + col*2 + (k&1)]
__global__ void prep_cls(const float* __restrict__ Wcls, float* __restrict__ Wp) {
  int g = blockIdx.x * blockDim.x + threadIdx.x;
  if (g >= H_ * VP_) return;
  int pr = g / (2 * VP_), rem = g - pr * (2 * VP_);
  int col = rem >> 1, kl = rem & 1;
  int k = 2 * pr + kl;
  Wp[g] = (col < V_) ? Wcls[(size_t)k * V_ + col] : 0.0f;
}

__global__ void prep_x0(const float* __restrict__ we, float* __restrict__ x) {
  int g = blockIdx.x * blockDim.x + threadIdx.x;
  if (g >= NROW * EPAD) return;
  int r = g / EPAD, k = g - r * EPAD;
  int b = r & 3;                                     // r = p*B_ + b
  x[g] = (k < E_) ? we[b * (T_ * E_) + k] : 0.0f;    // word_embs[b,0,k]
}

// ---------------- h0 = relu(obj @ W_map + b_map) ----------------
// wave tile: 32 rows x 64 cols ; grid 16 blocks x 8 waves
__global__ void h0_kernel(const float* __restrict__ obj, const float* __restrict__ Wmap,
                          const float* __restrict__ bmap, float* __restrict__ hA) {
  int lane = threadIdx.x & 31, wave = threadIdx.x >> 5;
  int wid = blockIdx.x * 8 + wave;                   // 128 tiles
  int rowBlk = wid >> 3, colBlk = wid & 7;
  int n16 = lane & 15, kh = lane >> 4;
  int row0 = rowBlk * 32, col0 = colBlk * 64;

  int srcrow[2];
#pragma unroll
  for (int mi = 0; mi < 2; ++mi) {
    int r = row0 + mi * 16 + n16;                    // r = p*B_ + b
    srcrow[mi] = (r & 3) * P_ + (r >> 2);            // obj_feats[b, p, :]
  }
  v8f acc[2][4] = {};
  for (int k0 = 0; k0 < F_; k0 += 4) {
    int ka = k0 + 2 * kh;
    v2f a[2];
#pragma unroll
    for (int mi = 0; mi < 2; ++mi)
      a[mi] = *(const v2f*)&obj[srcrow[mi] * F_ + ka];
    v2f bb[4];
#pragma unroll
    for (int ni = 0; ni < 4; ++ni) {
      int c = col0 + ni * 16 + n16;
      bb[ni].x = Wmap[ka * H_ + c];
      bb[ni].y = Wmap[(ka + 1) * H_ + c];
    }
#pragma unroll
    for (int mi = 0; mi < 2; ++mi)
#pragma unroll
      for (int ni = 0; ni < 4; ++ni)
        acc[mi][ni] = wmma4(a[mi], bb[ni], acc[mi][ni]);
  }
#pragma unroll
  for (int mi = 0; mi < 2; ++mi)
#pragma unroll
    for (int ni = 0; ni < 4; ++ni) {
      int c = col0 + ni * 16 + n16;
      float bc = bmap[c];
#pragma unroll
      for (int j = 0; j < 8; ++j) {
        int row = row0 + mi * 16 + 8 * kh + j;
        float v = acc[mi][ni][j] + bc;
        hA[row * H_ + c] = v > 0.0f ? v : 0.0f;
      }
    }
}

// ---------------- fused GRU cell ----------------
// wave tile: 16 rows x 32 cols of h' ; grid 64 blocks x 8 waves = 512 tiles
__global__ void gru_kernel(const float* __restrict__ x, const float* __restrict__ hin,
                           const float* __restrict__ WihT, const float* __restrict__ WhhT,
                           const float* __restrict__ b_ih, const float* __restrict__ b_hh,
                           float* __restrict__ hout, unsigned long long* __restrict__ amax) {
  int g = blockIdx.x * blockDim.x + threadIdx.x;
  if (g < NROW) amax[g] = 0ull;                      // reset argmax for this step

  int lane = threadIdx.x & 31, wave = threadIdx.x >> 5;
  int wid = blockIdx.x * 8 + wave;
  int rowBlk = wid >> 4, colBlk = wid & 15;
  int n16 = lane & 15, kh = lane >> 4;
  int row0 = rowBlk * 16, j0 = colBlk * 32;
  int arow = row0 + n16;

  v8f accR[2] = {}, accZ[2] = {}, accNI[2] = {}, accNH[2] = {};

  // gi = x @ W_ih^T  (K = EPAD, zero-padded)
  for (int k0 = 0; k0 < EPAD; k0 += 4) {
    int ka = k0 + 2 * kh;
    v2f a = *(const v2f*)&x[arow * EPAD + ka];
    int pb = (ka >> 1) * (2 * H3_);
#pragma unroll
    for (int s = 0; s < 2; ++s) {
      int c = j0 + s * 16 + n16;
      v2f bR = *(const v2f*)&WihT[pb + c * 2];
      v2f bZ = *(const v2f*)&WihT[pb + (c + H_) * 2];
      v2f bN = *(const v2f*)&WihT[pb + (c + 2 * H_) * 2];
      accR[s]  = wmma4(a, bR, accR[s]);
      accZ[s]  = wmma4(a, bZ, accZ[s]);
      accNI[s] = wmma4(a, bN, accNI[s]);
    }
  }
  // gh = h @ W_hh^T  (K = H_)
  for (int k0 = 0; k0 < H_; k0 += 4) {
    int ka = k0 + 2 * kh;
    v2f a = *(const v2f*)&hin[arow * H_ + ka];
    int pb = (ka >> 1) * (2 * H3_);
#pragma unroll
    for (int s = 0; s < 2; ++s) {
      int c = j0 + s * 16 + n16;
      v2f bR = *(const v2f*)&WhhT[pb + c * 2];
      v2f bZ = *(const v2f*)&WhhT[pb + (c + H_) * 2];
      v2f bN = *(const v2f*)&WhhT[pb + (c + 2 * H_) * 2];
      accR[s]  = wmma4(a, bR, accR[s]);
      accZ[s]  = wmma4(a, bZ, accZ[s]);
      accNH[s] = wmma4(a, bN, accNH[s]);
    }
  }
  // gates + h update, all in registers
#pragma unroll
  for (int s = 0; s < 2; ++s) {
    int c = j0 + s * 16 + n16;
    float bir  = b_ih[c] + b_hh[c];
    float biz  = b_ih[H_ + c] + b_hh[H_ + c];
    float bin_ = b_ih[2 * H_ + c];
    float bhn  = b_hh[2 * H_ + c];
#pragma unroll
    for (int j = 0; j < 8; ++j) {
      int row = row0 + 8 * kh + j;
      float rg = sigf(accR[s][j] + bir);
      float zg = sigf(accZ[s][j] + biz);
      float ng = tanhf(accNI[s][j] + bin_ + rg * (accNH[s][j] + bhn));
      float ho = hin[row * H_ + c];
      hout[row * H_ + c] = (1.0f - zg) * ng + zg * ho;
    }
  }
}

// ---------------- logits + transposed store + argmax ----------------
// block tile: 128 rows x 64 vocab cols (8 waves x 16 rows), W_cls K-chunks
// double-buffered in LDS via async global->LDS loads. grid 4*54 blocks.
__global__ void logits_kernel(const float* __restrict__ h, const float* __restrict__ Wp,
                              const float* __restrict__ bcls, float* __restrict__ out,
                              int t, unsigned long long* __restrict__ amax) {
  __shared__ __align__(16) float sB[2][KCH * 64];    // 2 x 8 KB

  int tid = threadIdx.x;
  int lane = tid & 31, wave = tid >> 5;
  int rowGrp = blockIdx.x / (VP_ / 64);              // 0..3
  int colBlk = blockIdx.x % (VP_ / 64);              // 0..53
  int n16 = lane & 15, kh = lane >> 4;
  int row0 = rowGrp * 128 + wave * 16;
  int col0 = colBlk * 64;
  int arow = row0 + n16;

  // staging assignment: 16 pair-rows x 128 floats per chunk
  int prow = tid >> 4, off = (tid & 15) * 8;
  const float* gbase = Wp + (size_t)prow * (2 * VP_) + col0 * 2 + off;

  // issue chunk 0   (chunk c covers pair-rows c*16 .. c*16+15)
  {
    float* l = &sB[0][prow * 128 + off];
    async_copy16(gbase, l);
    async_copy16(gbase + 4, l + 4);
  }

  v8f acc[4] = {};
  for (int c = 0; c < NCH; ++c) {
    if (c + 1 < NCH) {
      const float* gp = gbase + (size_t)(c + 1) * 16 * (2 * VP_);
      float* l = &sB[(c + 1) & 1][prow * 128 + off];
      async_copy16(gp, l);
      async_copy16(gp + 4, l + 4);
      async_wait2();
    } else {
      async_wait0();
    }
    __syncthreads();
    const float* Bb = sB[c & 1];
    int kbase = c * KCH;
#pragma unroll
    for (int kk = 0; kk < KCH; kk += 4) {
      int ka = kbase + kk + 2 * kh;
      v2f a = *(const v2f*)&h[arow * H_ + ka];
      int prel = (kk >> 1) + kh;
#pragma unroll
      for (int ni = 0; ni < 4; ++ni) {
        v2f b = *(const v2f*)&Bb[prel * 128 + (ni * 16 + n16) * 2];
        acc[ni] = wmma4(a, b, acc[ni]);
      }
    }
    __syncthreads();
  }

  unsigned long long best[8];
#pragma unroll
  for (int j = 0; j < 8; ++j) best[j] = 0ull;
#pragma unroll
  for (int ni = 0; ni < 4; ++ni) {
    int col = col0 + ni * 16 + n16;
    if (col >= V_) continue;
    float bc = bcls[col];
#pragma unroll
    for (int j = 0; j < 8; ++j) {
      int row = row0 + 8 * kh + j;
      float val = acc[ni][j] + bc;
      int p = row >> 2, b = row & 3;                 // r = p*B_ + b
      out[((size_t)((b * P_ + p) * STEPS + t)) * V_ + col] = val;
      unsigned long long pk = pack_key(val, col);
      if (pk > best[j]) best[j] = pk;
    }
  }
  // reduce each row's candidates across the 16 lanes of this half-wave
#pragma unroll
  for (int j = 0; j < 8; ++j) {
    unsigned long long v = best[j];
    for (int m = 1; m < 16; m <<= 1) {
      unsigned long long o = shfl_xor_u64(v, m);
      if (o > v) v = o;
    }
    if (n16 == 0) atomicMax(&amax[row0 + 8 * kh + j], v);
  }
}

// ---------------- greedy-token embedding gather ----------------
__global__ void gather_kernel(const unsigned long long* __restrict__ amax,
                              const float* __restrict__ emb, float* __restrict__ x) {
  int g = blockIdx.x * blockDim.x + threadIdx.x;
  if (g >= NROW * EPAD) return;
  int r = g / EPAD, k = g - r * EPAD;
  unsigned int idx = ~(unsigned int)amax[r];
  if (idx >= V_) idx = 0;
  x[g] = (k < E_) ? emb[(size_t)idx * E_ + k] : 0.0f;
}

extern "C" void kernel_launch(void* const* d_in, const int* in_sizes, int n_in,
                              void* d_out, int out_size, void* d_ws, size_t ws_size,
                              hipStream_t stream) {
  const float* word_embs = (const float*)d_in[0];
  const float* obj_feats = (const float*)d_in[1];
  const float* W_map     = (const float*)d_in[2];
  const float* b_map     = (const float*)d_in[3];
  const float* W_ih      = (const float*)d_in[4];
  const float* W_hh      = (const float*)d_in[5];
  const float* b_ih      = (const float*)d_in[6];
  const float* b_hh      = (const float*)d_in[7];
  const float* W_cls     = (const float*)d_in[8];
  const float* b_cls     = (const float*)d_in[9];
  const float* emb_table = (const float*)d_in[10];
  float* out = (float*)d_out;

  // workspace layout (floats): ~14.8 MB total
  float* ws   = (float*)d_ws;
  float* WihT = ws;                         // EPAD*H3_  (pair-interleaved)
  float* WhhT = WihT + EPAD * H3_;          // H_*H3_    (pair-interleaved)
  float* x    = WhhT + H_ * H3_;            // NROW*EPAD
  float* hA   = x + NROW * EPAD;            // NROW*H_
  float* hB   = hA + NROW * H_;             // NROW*H_
  float* Wp   = hB + NROW * H_;             // H_*VP_    (padded, pair-interleaved)
  unsigned long long* amax = (unsigned long long*)(Wp + H_ * VP_);  // NROW u64

  dim3 blk(256);
  prep_ih<<<(EPAD * H3_ + 255) / 256, blk, 0, stream>>>(W_ih, WihT);
  prep_hh<<<(H_ * H3_ + 255) / 256, blk, 0, stream>>>(W_hh, WhhT);
  prep_cls<<<(H_ * VP_ + 255) / 256, blk, 0, stream>>>(W_cls, Wp);
  prep_x0<<<(NROW * EPAD + 255) / 256, blk, 0, stream>>>(word_embs, x);
  h0_kernel<<<16, blk, 0, stream>>>(obj_feats, W_map, b_map, hA);

  for (int t = 0; t < STEPS; ++t) {
    const float* hin = (t & 1) ? hB : hA;
    float* hout      = (t & 1) ? hA : hB;
    gru_kernel<<<64, blk, 0, stream>>>(x, hin, WihT, WhhT, b_ih, b_hh, hout, amax);
    logits_kernel<<<4 * (VP_ / 64), blk, 0, stream>>>(hout, Wp, b_cls, out, t, amax);
    gather_kernel<<<(NROW * EPAD + 255) / 256, blk, 0, stream>>>(amax, emb_table, x);
  }
}